// NMTModel_17652315587030
// MI455X (gfx1250) — compile-verified
//
#include <hip/hip_runtime.h>

// Problem constants (from reference): B=8, T=100, S=400, K=8, H=512, V=32000
#define Bd 8
#define Td 100
#define Sd 400
#define Kd 8
#define Hd 512
#define Vd 32000
#define Md (Bd * Td)   // 800 output rows

typedef __attribute__((ext_vector_type(2))) float v2f;
typedef __attribute__((ext_vector_type(8))) float v8f;

// ---------------------------------------------------------------------------
// K1: logits = dec_out @ W + bias   (exact fp32 via V_WMMA_F32_16X16X4_F32)
// grid = (Vd/256, Md/32), block = 256 threads (8 wave32).
// Each block: 32 rows x 256 cols. Each wave: 2 M-tiles x 2 N-tiles (4 v8f
// accumulators) -> 4 WMMAs per 4 B-loads (halves W L2 traffic vs 16-row tile).
// A tile is streamed K-chunk by K-chunk (KT=128) into double-buffered LDS via
// GLOBAL_LOAD_ASYNC_TO_LDS_B128 (ASYNCcnt), overlapped with WMMA compute.
// ---------------------------------------------------------------------------
#define KT    128            // K columns staged per chunk
#define NCH   (Hd / KT)      // 4 chunks
#define LDAC  132            // chunk row stride (floats): 132 % 64 == 4 -> conflict-free

__device__ __forceinline__ void issue_a_chunk(const float* __restrict__ A,
                                              int mbase, int ks,
                                              float* buf, int tid)
{
    // 32 rows x 128 floats = 1024 x b128 transfers, 4 per thread, coalesced.
    #pragma unroll
    for (int t = 0; t < 4; ++t) {
        const int j  = tid + t * 256;
        const int r  = j >> 5;        // 0..31
        const int cq = j & 31;        // b128 index within row
        const float* g = A + (size_t)(mbase + r) * Hd + ks + cq * 4;
        const unsigned l = (unsigned)(size_t)(buf + r * LDAC + cq * 4);
        asm volatile("global_load_async_to_lds_b128 %0, %1, off"
                     :: "v"(l), "v"(g) : "memory");
    }
}

__global__ __launch_bounds__(256) void gemm_logits_kernel(
    const float* __restrict__ A,     // (Md, Hd) dec_out flattened
    const float* __restrict__ W,     // (Hd, Vd)
    const float* __restrict__ bias,  // (Vd)
    float* __restrict__ logits)      // (Md, Vd)
{
    __shared__ float ldsA[2][32 * LDAC];   // 2 x 16.9 KB double buffer

    const int tid   = threadIdx.x;
    const int wave  = tid >> 5;
    const int lane  = tid & 31;
    const int lid   = lane & 15;     // A row within tile / B col within tile
    const int lhalf = lane >> 4;     // selects K pair {0,1} vs {2,3}
    const int mbase = blockIdx.y * 32;
    const int nbase = blockIdx.x * 256;

    const int col0 = nbase + wave * 32 + lid;
    const int col1 = col0 + 16;

    // Prime the async pipeline: chunks 0 and 1 in flight.
    issue_a_chunk(A, mbase, 0,  ldsA[0], tid);
    issue_a_chunk(A, mbase, KT, ldsA[1], tid);

    v8f acc00 = {};   // M-tile 0 x N-tile 0
    v8f acc01 = {};   // M-tile 0 x N-tile 1
    v8f acc10 = {};   // M-tile 1 x N-tile 0
    v8f acc11 = {};   // M-tile 1 x N-tile 1

    for (int c = 0; c < NCH; ++c) {
        // Wait for chunk c's 4 per-wave async issues (next chunk may stay in flight).
        if (c < NCH - 1)
            asm volatile("s_wait_asynccnt 0x4" ::: "memory");
        else
            asm volatile("s_wait_asynccnt 0x0" ::: "memory");
        __syncthreads();

        const float2* a0 = (const float2*)&ldsA[c & 1][lid * LDAC];
        const float2* a1 = (const float2*)&ldsA[c & 1][(16 + lid) * LDAC];
        const int ksb = c * KT;

        #pragma unroll 4
        for (int kb = 0; kb < KT; kb += 4) {
            const int kk = kb + 2 * lhalf;       // this lane's K pair in chunk
            const float2 af0 = a0[kk >> 1];
            const float2 af1 = a1[kk >> 1];
            v2f va0; va0.x = af0.x; va0.y = af0.y;
            v2f va1; va1.x = af1.x; va1.y = af1.y;

            const int wk = ksb + kk;             // global K for B fragment
            v2f b0; b0.x = W[(size_t)wk * Vd + col0]; b0.y = W[(size_t)(wk + 1) * Vd + col0];
            v2f b1; b1.x = W[(size_t)wk * Vd + col1]; b1.y = W[(size_t)(wk + 1) * Vd + col1];

            acc00 = __builtin_amdgcn_wmma_f32_16x16x4_f32(false, va0, false, b0,
                                                          (short)0, acc00, false, false);
            acc01 = __builtin_amdgcn_wmma_f32_16x16x4_f32(false, va0, false, b1,
                                                          (short)0, acc01, false, false);
            acc10 = __builtin_amdgcn_wmma_f32_16x16x4_f32(false, va1, false, b0,
                                                          (short)0, acc10, false, false);
            acc11 = __builtin_amdgcn_wmma_f32_16x16x4_f32(false, va1, false, b1,
                                                          (short)0, acc11, false, false);
        }
        __syncthreads();                      // everyone done reading buf[c&1]
        if (c + 2 < NCH)
            issue_a_chunk(A, mbase, (c + 2) * KT, ldsA[c & 1], tid);
    }

    // Epilogue: bias add + store. C/D layout: VGPR r -> M = r + 8*lhalf, N = lid.
    const float bb0 = bias[col0];
    const float bb1 = bias[col1];
    #pragma unroll
    for (int r = 0; r < 8; ++r) {
        const int grow0 = mbase + r + 8 * lhalf;       // M-tile 0
        const int grow1 = grow0 + 16;                  // M-tile 1
        logits[(size_t)grow0 * Vd + col0] = acc00[r] + bb0;
        logits[(size_t)grow0 * Vd + col1] = acc01[r] + bb1;
        logits[(size_t)grow1 * Vd + col0] = acc10[r] + bb0;
        logits[(size_t)grow1 * Vd + col1] = acc11[r] + bb1;
    }
}

// ---------------------------------------------------------------------------
// K2: per output row (b,t): online softmax stats over logits row, then the
// sparse copy-distribution scatter (S*K = 3200 ds_add_f32 into an LDS vocab
// window) fused with the final log-blend. Avoids materializing the 410 MB
// (B,S,V) scatter tensor and the 20.5 GFLOP dense blend GEMM entirely.
// grid = Md blocks, block = 256 threads. LDS window 8000 floats (32 KB).
// ---------------------------------------------------------------------------
#define VT 8000   // vocab window; 4 windows cover V=32000

__global__ __launch_bounds__(256) void blend_kernel(
    const float* __restrict__ logits,       // (Md, Vd)
    const float* __restrict__ weights,      // (B, T, S)
    const float* __restrict__ p_trans,      // (B, T, 1)
    const float* __restrict__ trans_probs,  // (B, S, K)
    const float* __restrict__ probs,        // (B, S, K)
    const int*   __restrict__ idxes,        // (B, S, K)
    float* __restrict__ out)                // (B, T, V)
{
    __shared__ float vtab[VT];
    __shared__ float rm[256];
    __shared__ float rs[256];

    const int row = blockIdx.x;            // 0..799 == b*T + t
    const int b   = row / Td;
    const int tid = threadIdx.x;
    const float* lrow = logits + (size_t)row * Vd;

    // --- Phase 1: online (max, sum-exp) over the 32000-wide logits row ---
    float m = -3.4e38f, s = 0.0f;
    for (int v = tid; v < Vd; v += 256) {
        const float x  = lrow[v];
        const float nm = fmaxf(m, x);
        s = s * __expf(m - nm) + __expf(x - nm);
        m = nm;
    }
    rm[tid] = m; rs[tid] = s;
    __syncthreads();
    for (int off = 128; off > 0; off >>= 1) {
        if (tid < off) {
            const float m2 = rm[tid + off], s2 = rs[tid + off];
            const float nm = fmaxf(rm[tid], m2);
            rs[tid] = rs[tid] * __expf(rm[tid] - nm) + s2 * __expf(m2 - nm);
            rm[tid] = nm;
        }
        __syncthreads();
    }
    const float M   = rm[0];
    const float inv = 1.0f / rs[0];
    const float pt  = p_trans[row];
    const float q   = 1.0f - pt;
    const float* wrow = weights + (size_t)row * Sd;

    // --- Phases 2+3 per vocab window: zero, scatter, blend+store ---
    for (int vbase = 0; vbase < Vd; vbase += VT) {
        for (int v = tid; v < VT; v += 256) vtab[v] = 0.0f;
        __syncthreads();

        for (int i = tid; i < Sd * Kd; i += 256) {
            const int si  = i >> 3;
            const int off = (b * Sd + si) * Kd + (i & 7);
            const int vid = idxes[off] - vbase;
            if ((unsigned)vid < (unsigned)VT) {
                const float p  = probs[off];
                const float tp = (p > 0.05f) ? trans_probs[off] : 0.0f;
                atomicAdd(&vtab[vid], wrow[si] * tp);   // ds_add_f32
            }
        }
        __syncthreads();

        for (int v = tid; v < VT; v += 256) {
            const int gv   = vbase + v;
            const float sc = __expf(lrow[gv] - M) * inv;   // softmax prob
            out[(size_t)row * Vd + gv] = __logf(pt * vtab[v] + q * sc);
        }
        __syncthreads();
    }
}

// ---------------------------------------------------------------------------
extern "C" void kernel_launch(void* const* d_in, const int* in_sizes, int n_in,
                              void* d_out, int out_size, void* d_ws, size_t ws_size,
                              hipStream_t stream)
{
    const float* dec_out     = (const float*)d_in[0];  // (B,T,H)
    const float* W           = (const float*)d_in[1];  // (H,V)
    const float* bias        = (const float*)d_in[2];  // (V)
    const float* weights     = (const float*)d_in[3];  // (B,T,S)
    const float* p_trans     = (const float*)d_in[4];  // (B,T,1)
    const float* trans_probs = (const float*)d_in[5];  // (B,S,K)
    const float* probs       = (const float*)d_in[6];  // (B,S,K)
    const int*   idxes       = (const int*)d_in[7];    // (B,S,K)
    float* out = (float*)d_out;

    float* logits = (float*)d_ws;   // Md * Vd * 4 = 102.4 MB scratch

    dim3 g1(Vd / 256, Md / 32);     // (125, 25)
    gemm_logits_kernel<<<g1, 256, 0, stream>>>(dec_out, W, bias, logits);

    blend_kernel<<<Md, 256, 0, stream>>>(logits, weights, p_trans,
                                         trans_probs, probs, idxes, out);
}